// GRUStateEstimation_77111842832531
// MI455X (gfx1250) — compile-verified
//
#include <hip/hip_runtime.h>
#include <hip/hip_bf16.h>
#include <math.h>

typedef float v2f __attribute__((ext_vector_type(2)));
typedef float v8f __attribute__((ext_vector_type(8)));

#define GRU_B   256
#define GRU_T   200
#define GRU_IN  128
#define GRU_H   256
#define GRU_G   768   // 3*H
#define GRU_OUT 118
#define NT_ALL  48    // G/16 n-tiles
#define HB_STRIDE 260 // 256 + 4 pad: conflict-free ds_load_b64 A-fragments

__device__ __forceinline__ float sigmoidf_fast(float x) {
    return 1.0f / (1.0f + __expf(-x));
}

// ---------------------------------------------------------------------------
// Pack W[G rows][K cols] (the [3H, K] weight) into WMMA B-fragment order:
// for k-block kb and n-tile nt, lane L holds float2 = W[nt*16+row][kb*4+2*half + {0,1}]
// at Bp[( (kb*48 + nt)*32 + L )]. Inner-loop B loads become ONE coalesced
// global_load_b64 per fragment (wave reads a contiguous 256B line).
// ---------------------------------------------------------------------------
__global__ void pack_b_kernel(const float* __restrict__ W, float2* __restrict__ Bp,
                              int K) {
    int idx = blockIdx.x * blockDim.x + threadIdx.x;   // over (K/4)*48*32
    int total = (K / 4) * NT_ALL * 32;
    if (idx >= total) return;
    int lane = idx & 31;
    int tmp  = idx >> 5;
    int nt   = tmp % NT_ALL;
    int kb   = tmp / NT_ALL;
    int half = lane >> 4, row = lane & 15;
    int n = nt * 16 + row;
    int k = kb * 4 + 2 * half;
    float2 v;
    v.x = W[(size_t)n * K + k];
    v.y = W[(size_t)n * K + k + 1];
    Bp[idx] = v;
}

// ---------------------------------------------------------------------------
// gx = A[M,K] @ W.T + bias  (input projection). One block per 16-row M-tile;
// wave w owns 6 consecutive n-tiles, so each A fragment is loaded once per
// k-iter and reused across 6 WMMAs (6 independent accumulator chains).
// ---------------------------------------------------------------------------
__global__ __launch_bounds__(256) void gx_gemm_kernel(
    const float* __restrict__ A, const float2* __restrict__ Bp,
    const float* __restrict__ bias, float* __restrict__ Cout,
    int K) {
    const int G = GRU_G;
    int wave = threadIdx.x >> 5;
    int lane = threadIdx.x & 31;
    int half = lane >> 4, row = lane & 15;
    int m0 = blockIdx.x * 16;
    int ntbase = wave * 6;

    v8f acc[6] = {};
    for (int kb = 0; kb < K / 4; ++kb) {
        v2f a = *(const v2f*)(A + (size_t)(m0 + row) * K + kb * 4 + 2 * half);
#pragma unroll
        for (int i = 0; i < 6; ++i) {
            float2 bv = Bp[((size_t)kb * NT_ALL + ntbase + i) * 32 + lane];
            v2f b; b.x = bv.x; b.y = bv.y;
            acc[i] = __builtin_amdgcn_wmma_f32_16x16x4_f32(
                false, a, false, b, (short)0, acc[i], false, false);
        }
    }
#pragma unroll
    for (int i = 0; i < 6; ++i) {
        int n0 = (ntbase + i) * 16;
#pragma unroll
        for (int vr = 0; vr < 8; ++vr) {
            int m = m0 + vr + 8 * half;
            int n = n0 + row;
            Cout[(size_t)m * G + n] = acc[i][vr] + bias[n];
        }
    }
}

// ---------------------------------------------------------------------------
// Recurrent GRU layer. 16 persistent workgroups, each owns 16 batch rows for
// all T steps. h lives in LDS; per step: gh = h @ Whh.T via f32 WMMA
// (A from LDS, fragment-packed B streamed from L2 front-to-back), then fused
// sigmoid/tanh gate update. Wave w owns hidden cols [w*32, w*32+32) across
// all three gate regions (n-tiles g*16 + w*2 + c), so r,z,n are
// register-resident for the update.
// ---------------------------------------------------------------------------
__global__ __launch_bounds__(256) void gru_recurrent_kernel(
    const float* __restrict__ gx,      // [B,T,G] precomputed x@Wih.T + bih
    const float2* __restrict__ Bp,     // fragment-packed Whh
    const float* __restrict__ bhh,     // [G]
    float* __restrict__ hseq,          // [B,T,H] or nullptr
    float* __restrict__ hlast) {       // [B,H]  or nullptr
    const int G = GRU_G, H = GRU_H, T = GRU_T;
    __shared__ __align__(16) float hbuf[16 * HB_STRIDE];

    int b0   = blockIdx.x * 16;
    int wave = threadIdx.x >> 5;
    int lane = threadIdx.x & 31;
    int half = lane >> 4, row = lane & 15;
    int jbase = wave * 32;

    for (int i = threadIdx.x; i < 16 * HB_STRIDE; i += 256) hbuf[i] = 0.0f;
    __syncthreads();

    for (int t = 0; t < T; ++t) {
        // hint this step's gx rows toward the caches while the GEMM runs
        __builtin_prefetch(gx + ((size_t)(b0 + row) * T + t) * G + jbase + half * 384, 0, 0);

        v8f acc[3][2];
#pragma unroll
        for (int g = 0; g < 3; ++g)
#pragma unroll
            for (int c = 0; c < 2; ++c) { v8f z = {}; acc[g][c] = z; }

        for (int kb = 0; kb < H / 4; ++kb) {
            v2f a = *(const v2f*)(&hbuf[row * HB_STRIDE + kb * 4 + 2 * half]);
#pragma unroll
            for (int g = 0; g < 3; ++g) {
#pragma unroll
                for (int c = 0; c < 2; ++c) {
                    int nt = g * 16 + wave * 2 + c;
                    float2 bv = Bp[((size_t)kb * NT_ALL + nt) * 32 + lane];
                    v2f b; b.x = bv.x; b.y = bv.y;
                    acc[g][c] = __builtin_amdgcn_wmma_f32_16x16x4_f32(
                        false, a, false, b, (short)0, acc[g][c], false, false);
                }
            }
        }
        __syncthreads();  // all A-fragment reads of hbuf are done

#pragma unroll
        for (int c = 0; c < 2; ++c) {
            int j = jbase + c * 16 + row;
            float bh_r = bhh[j];
            float bh_z = bhh[256 + j];
            float bh_n = bhh[512 + j];
#pragma unroll
            for (int vr = 0; vr < 8; ++vr) {
                int m = vr + 8 * half;  // batch row within tile
                const float* gxp = gx + ((size_t)(b0 + m) * T + t) * G;
                float r  = sigmoidf_fast(gxp[j]       + acc[0][c][vr] + bh_r);
                float z  = sigmoidf_fast(gxp[256 + j] + acc[1][c][vr] + bh_z);
                float hn = tanhf(gxp[512 + j] + r * (acc[2][c][vr] + bh_n));
                float hold = hbuf[m * HB_STRIDE + j];
                float hnew = (1.0f - z) * hn + z * hold;
                hbuf[m * HB_STRIDE + j] = hnew;  // own columns only: no x-wave hazard
                if (hseq)
                    hseq[((size_t)(b0 + m) * T + t) * H + j] = hnew;
            }
        }
        __syncthreads();  // writes visible before next step's A reads
    }

    if (hlast) {
        for (int i = threadIdx.x; i < 16 * H; i += 256) {
            int m = i / H, j = i % H;
            hlast[(size_t)(b0 + m) * H + j] = hbuf[m * HB_STRIDE + j];
        }
    }
}

// ---------------------------------------------------------------------------
// out[b][o] = hlast[b] . fcW[o] + fcb[o]   (tiny: 7.7 MFLOP, plain VALU)
// ---------------------------------------------------------------------------
__global__ void fc_kernel(const float* __restrict__ h, const float* __restrict__ W,
                          const float* __restrict__ bias, float* __restrict__ out) {
    int i = blockIdx.x * blockDim.x + threadIdx.x;
    if (i >= GRU_B * GRU_OUT) return;
    int b = i / GRU_OUT, o = i % GRU_OUT;
    const float* hp = h + (size_t)b * GRU_H;
    const float* wp = W + (size_t)o * GRU_H;
    float s = bias[o];
#pragma unroll 4
    for (int k = 0; k < GRU_H; ++k) s += hp[k] * wp[k];
    out[i] = s;
}

// ---------------------------------------------------------------------------
extern "C" void kernel_launch(void* const* d_in, const int* in_sizes, int n_in,
                              void* d_out, int out_size, void* d_ws, size_t ws_size,
                              hipStream_t stream) {
    const float* x    = (const float*)d_in[0];
    const float* Wih0 = (const float*)d_in[1];
    const float* Whh0 = (const float*)d_in[2];
    const float* bih0 = (const float*)d_in[3];
    const float* bhh0 = (const float*)d_in[4];
    const float* Wih1 = (const float*)d_in[5];
    const float* Whh1 = (const float*)d_in[6];
    const float* bih1 = (const float*)d_in[7];
    const float* bhh1 = (const float*)d_in[8];
    const float* fcW  = (const float*)d_in[9];
    const float* fcb  = (const float*)d_in[10];
    float* out = (float*)d_out;

    const size_t M = (size_t)GRU_B * GRU_T;       // 51200
    float* ws     = (float*)d_ws;
    float* BihP   = ws;                           // packed Wih: up to 256*768
    float* BhhP   = BihP + (size_t)256 * GRU_G;   // packed Whh: 256*768
    float* gx     = BhhP + (size_t)256 * GRU_G;   // M*768   (reused by both layers)
    float* hseq   = gx   + M * GRU_G;             // M*256
    float* hlast  = hseq + M * GRU_H;             // 256*256

    const int TPB = 256;
    dim3 blk(TPB);
    int gemmGrid = (int)(M / 16);                 // 3200 blocks, 48 tiles each

    // ---------------- layer 0 ----------------
    pack_b_kernel<<<(GRU_IN * 384 + TPB - 1) / TPB, blk, 0, stream>>>(
        Wih0, (float2*)BihP, GRU_IN);
    pack_b_kernel<<<(GRU_H * 384 + TPB - 1) / TPB, blk, 0, stream>>>(
        Whh0, (float2*)BhhP, GRU_H);
    gx_gemm_kernel<<<gemmGrid, blk, 0, stream>>>(x, (const float2*)BihP, bih0, gx, GRU_IN);
    gru_recurrent_kernel<<<GRU_B / 16, blk, 0, stream>>>(
        gx, (const float2*)BhhP, bhh0, hseq, nullptr);

    // ---------------- layer 1 ----------------
    pack_b_kernel<<<(GRU_H * 384 + TPB - 1) / TPB, blk, 0, stream>>>(
        Wih1, (float2*)BihP, GRU_H);
    pack_b_kernel<<<(GRU_H * 384 + TPB - 1) / TPB, blk, 0, stream>>>(
        Whh1, (float2*)BhhP, GRU_H);
    gx_gemm_kernel<<<gemmGrid, blk, 0, stream>>>(hseq, (const float2*)BihP, bih1, gx, GRU_H);
    gru_recurrent_kernel<<<GRU_B / 16, blk, 0, stream>>>(
        gx, (const float2*)BhhP, bhh1, nullptr, hlast);

    // ---------------- FC head ----------------
    fc_kernel<<<(GRU_B * GRU_OUT + TPB - 1) / TPB, blk, 0, stream>>>(hlast, fcW, fcb, out);
}